// BRB_28776280883405
// MI455X (gfx1250) — compile-verified
//
#include <hip/hip_runtime.h>
#include <hip/hip_bf16.h>
#include <math.h>

#define BATCH 512
#define RULE  2048
#define ATT   256
#define RES   64
#define KP    (2*ATT)          // packed K = 512 : (x, x^2) interleaved
#define EPS   1e-10f

typedef float v2f __attribute__((ext_vector_type(2)));
typedef float v8f __attribute__((ext_vector_type(8)));

typedef unsigned int uint32x4 __attribute__((ext_vector_type(4)));
typedef int          int32x4  __attribute__((ext_vector_type(4)));
typedef int          int32x8  __attribute__((ext_vector_type(8)));

// ---------------------------------------------------------------------------
// Kernel 1: pack rules:  RP[j, 2k] = -2*d^2*a,  RP[j, 2k+1] = d^2
//           CJ[j] = sum_k d^2 * a^2   (LDS tree reduction)
// ---------------------------------------------------------------------------
__global__ void prep_rules(const float* __restrict__ att,
                           const float* __restrict__ dis,
                           float* __restrict__ RP_, float* __restrict__ CJ) {
    __shared__ float red[256];
    const int j = blockIdx.x;
    const int k = threadIdx.x;
    const float a  = att[j * ATT + k];
    const float d  = dis[j * ATT + k];
    const float d2 = d * d;
    RP_[j * KP + 2 * k]     = -2.0f * d2 * a;
    RP_[j * KP + 2 * k + 1] = d2;
    red[k] = d2 * a * a;
    __syncthreads();
    for (int off = 128; off > 0; off >>= 1) {
        if (k < off) red[k] += red[k + off];
        __syncthreads();
    }
    if (k == 0) CJ[j] = red[0];
}

// ---------------------------------------------------------------------------
// Kernel 2: pack inputs:  XP[i, 2k] = x,  XP[i, 2k+1] = x^2
// ---------------------------------------------------------------------------
__global__ void prep_inputs(const float* __restrict__ x, float* __restrict__ XP_) {
    const int i = blockIdx.x;
    const int k = threadIdx.x;
    const float v = x[i * ATT + k];
    XP_[i * KP + 2 * k]     = v;
    XP_[i * KP + 2 * k + 1] = v * v;
}

// ---------------------------------------------------------------------------
// Kernel 3: 2-way softmax of res along last axis
// ---------------------------------------------------------------------------
__global__ void softmax2(const float* __restrict__ res, float* __restrict__ SM_) {
    const int t  = blockIdx.x * blockDim.x + threadIdx.x;   // < RULE*RES
    const float r0 = res[2 * t];
    const float r1 = res[2 * t + 1];
    const float m  = fmaxf(r0, r1);
    const float e0 = expf(r0 - m);
    const float e1 = expf(r1 - m);
    const float is = 1.0f / (e0 + e1);
    SM_[2 * t]     = e0 * is;
    SM_[2 * t + 1] = e1 * is;
}

// ---------------------------------------------------------------------------
// Kernel 4: fp32 WMMA GEMM  S = CJ + XP(512x512) * RP^T(512x2048),
//           then W = exp(-S).
//
// All 8 waves of a block share the same 16 batch rows (A tile, 32 KB).
// The A tile is DMA'd into LDS once per block via the Tensor Data Mover
// (tensor_load_to_lds, TENSORcnt), then each wave feeds its WMMA A-operand
// from LDS (ds_load_b64 broadcasts) while B streams from global.
// ---------------------------------------------------------------------------
__global__ void __launch_bounds__(256)
gemm_w(const float* __restrict__ XP_, const float* __restrict__ RP_,
       const float* __restrict__ CJ, float* __restrict__ W_) {
    __shared__ float lds_a[16 * KP];          // 16 rows x 512 floats = 32 KB

    const int wave = threadIdx.x >> 5;
    const int lane = threadIdx.x & 31;
    const int tile = blockIdx.x * 8 + wave;   // 4096 tiles = 32 x 128
    const int mt   = tile >> 7;               // 0..31  (batch tile) — same for all 8 waves
    const int nt   = tile & 127;              // 0..127 (rule tile)
    const int row  = lane & 15;
    const int kh   = lane >> 4;

    // ---- stage A tile (XP rows mt*16 .. mt*16+15) into LDS ------------------
#if __has_builtin(__builtin_amdgcn_tensor_load_to_lds)
    if (threadIdx.x == 0) {
        const unsigned long long gaddr =
            (unsigned long long)(XP_ + (unsigned long long)mt * 16ull * KP);
        const unsigned lds_addr = (unsigned)(unsigned long long)(&lds_a[0]);

        // D# group 0: count=1 | lds_addr | global_addr(57b) | type=2
        uint32x4 g0;
        g0[0] = 1u;                                        // count=1, user mode
        g0[1] = lds_addr;                                  // LDS byte address
        g0[2] = (unsigned)(gaddr & 0xFFFFFFFFull);         // global_addr[31:0]
        g0[3] = (unsigned)((gaddr >> 32) & 0x1FFFFFFull)   // global_addr[56:32]
              | (2u << 30);                                // type = 2 ("image")

        // D# group 1: data_size=4B, tensor 512x16, tile 512x16, stride0=512
        int32x8 g1;
        g1[0] = (2 << 16);                 // workgroup_mask=0, data_size=2 (4 bytes)
        g1[1] = (int)((KP & 0xFFFF) << 16);                // tensor_dim0[15:0]=512
        g1[2] = (int)(((KP >> 16) & 0xFFFF)                // tensor_dim0[31:16]
              | ((16 & 0xFFFF) << 16));                    // tensor_dim1[15:0]=16
        g1[3] = (int)(((16 >> 16) & 0xFFFF)                // tensor_dim1[31:16]
              | ((KP & 0xFFFF) << 16));                    // tile_dim0=512
        g1[4] = 16;                                        // tile_dim1=16, tile_dim2=0
        g1[5] = KP;                                        // tensor_dim0_stride[31:0]=512
        g1[6] = 0;                                         // stride[47:32] | dim1_stride lo
        g1[7] = 0;

        int32x4 gz = {0, 0, 0, 0};                         // groups 2/3: unused (2-D)
#if defined(__clang_major__) && (__clang_major__ >= 23)
        int32x8 gz8 = {0, 0, 0, 0, 0, 0, 0, 0};
        __builtin_amdgcn_tensor_load_to_lds(g0, g1, gz, gz, gz8, 0);
#else
        __builtin_amdgcn_tensor_load_to_lds(g0, g1, gz, gz, 0);
#endif
        __builtin_amdgcn_s_wait_tensorcnt(0);              // TENSORcnt is per-wave
    }
    __syncthreads();
#else
    for (int t = threadIdx.x; t < 16 * KP; t += 256)
        lds_a[t] = XP_[(unsigned long long)mt * 16ull * KP + t];
    __syncthreads();
#endif

    // ---- K loop: A from LDS, B from global ---------------------------------
    const float* la = lds_a + row * KP + 2 * kh;
    const float* rr = RP_ + (nt * 16 + row) * KP + 2 * kh;

    v8f acc = {};
    for (int kb = 0; kb < KP; kb += 64) {
        __builtin_prefetch(rr + kb + 64, 0, 1);            // -> global_prefetch_b8
        #pragma unroll
        for (int k = kb; k < kb + 64; k += 4) {
            v2f a = *(const v2f*)(la + k);                 // ds_load_b64
            v2f b = *(const v2f*)(rr + k);                 // global_load_b64
            acc = __builtin_amdgcn_wmma_f32_16x16x4_f32(
                false, a, false, b, (short)0, acc, false, false);
        }
    }

    // C/D layout: lane L: N = L%16 ; VGPR v holds M = v + 8*(L/16)
    const int j  = nt * 16 + row;
    const float cj = CJ[j];
    #pragma unroll
    for (int v = 0; v < 8; ++v) {
        const int i = mt * 16 + v + 8 * kh;
        W_[i * RULE + j] = expf(-(acc[v] + cj));
    }
}

// ---------------------------------------------------------------------------
// Kernel 5/6: scalar prod over all (1 - w)   (matches jnp.prod(md))
// ---------------------------------------------------------------------------
__global__ void partial_prod(const float* __restrict__ W_, float* __restrict__ part) {
    __shared__ float s[256];
    const int i = blockIdx.x;
    const int t = threadIdx.x;
    float p = 1.0f;
    for (int j = t; j < RULE; j += 256) p *= (1.0f - W_[i * RULE + j]);
    s[t] = p;
    __syncthreads();
    for (int off = 128; off > 0; off >>= 1) {
        if (t < off) s[t] *= s[t + off];
        __syncthreads();
    }
    if (t == 0) part[i] = s[0];
}

__global__ void final_prod(const float* __restrict__ part, float* __restrict__ P) {
    __shared__ float s[256];
    const int t = threadIdx.x;
    s[t] = part[t] * part[t + 256];
    __syncthreads();
    for (int off = 128; off > 0; off >>= 1) {
        if (t < off) s[t] *= s[t + off];
        __syncthreads();
    }
    if (t == 0) *P = s[0];
}

// ---------------------------------------------------------------------------
// Kernel 7: bc product.  One block per batch row; w-row staged in LDS and
// broadcast to all 64 res-lanes; SM read coalesced (b64 per lane).
// out[i,r] = log((prod1 - P + EPS) / (prod0 - P + EPS))
// ---------------------------------------------------------------------------
__global__ void bc_kernel(const float* __restrict__ W_, const float* __restrict__ SM_,
                          const float* __restrict__ P, float* __restrict__ out) {
    __shared__ float wrow[RULE];
    const int i = blockIdx.x;
    const int r = threadIdx.x;                 // 0..63
    for (int j = r; j < RULE; j += 64) wrow[j] = W_[i * RULE + j];
    __syncthreads();

    float p0 = 1.0f, p1 = 1.0f;
    #pragma unroll 4
    for (int j = 0; j < RULE; ++j) {
        const float w  = wrow[j];
        const v2f   q  = *(const v2f*)(SM_ + (j * RES + r) * 2);
        p0 *= fmaf(w, q.x - 1.0f, 1.0f);
        p1 *= fmaf(w, q.y - 1.0f, 1.0f);
    }
    const float Pm = *P;
    const float b0 = p0 - Pm + EPS;
    const float b1 = p1 - Pm + EPS;
    out[i * RES + r] = logf(b1 / b0);
}

// ---------------------------------------------------------------------------
extern "C" void kernel_launch(void* const* d_in, const int* in_sizes, int n_in,
                              void* d_out, int out_size, void* d_ws, size_t ws_size,
                              hipStream_t stream) {
    const float* inputs = (const float*)d_in[0];   // (512, 256)
    const float* att    = (const float*)d_in[1];   // (2048, 256)
    const float* dis    = (const float*)d_in[2];   // (2048, 256)
    const float* res    = (const float*)d_in[3];   // (2048, 64, 2)

    float* ws   = (float*)d_ws;
    float* RP_  = ws;                       // RULE*KP      = 1048576 f
    float* XP_  = RP_  + RULE * KP;         // BATCH*KP     =  262144 f
    float* CJ   = XP_  + BATCH * KP;        // RULE         =    2048 f
    float* SM_  = CJ   + RULE;              // RULE*RES*2   =  262144 f
    float* W_   = SM_  + RULE * RES * 2;    // BATCH*RULE   = 1048576 f
    float* PART = W_   + BATCH * RULE;      // BATCH        =     512 f
    float* P    = PART + BATCH;             // 1 f          (~10.5 MB total)

    prep_rules  <<<RULE, 256, 0, stream>>>(att, dis, RP_, CJ);
    prep_inputs <<<BATCH, 256, 0, stream>>>(inputs, XP_);
    softmax2    <<<(RULE * RES) / 256, 256, 0, stream>>>(res, SM_);
    gemm_w      <<<(BATCH / 16) * (RULE / 16) / 8, 256, 0, stream>>>(XP_, RP_, CJ, W_);
    partial_prod<<<BATCH, 256, 0, stream>>>(W_, PART);
    final_prod  <<<1, 256, 0, stream>>>(PART, P);
    bc_kernel   <<<BATCH, RES, 0, stream>>>(W_, SM_, P, (float*)d_out);
}